// ModernMPNN_28441273434169
// MI455X (gfx1250) — compile-verified
//
#include <hip/hip_runtime.h>
#include <hip/hip_bf16.h>

typedef __attribute__((ext_vector_type(2))) float v2f;
typedef __attribute__((ext_vector_type(8))) float v8f;

#define NND 131072
#define NE  262144
#define NG  4096
#define CH  256
#define EPSV 1e-5f

// ---------------- utility kernels ----------------

// n must be a multiple of 4 (all our buffers are)
__global__ void fill4_kernel(float4* __restrict__ p, float v, long long n4) {
    long long i = (long long)blockIdx.x * blockDim.x + threadIdx.x;
    if (i < n4) p[i] = make_float4(v, v, v, v);
}

__global__ void degree_kernel(const int* __restrict__ dst, float* __restrict__ deg, int E) {
    int i = blockIdx.x * blockDim.x + threadIdx.x;
    if (i < E) atomicAdd(&deg[dst[i]], 1.0f);
}

__global__ void rsqrt_kernel(float* __restrict__ p, int n) {
    int i = blockIdx.x * blockDim.x + threadIdx.x;
    if (i < n) p[i] = rsqrtf(p[i]);
}

// ---------------- WMMA fp32 GEMM: out[N x 256] = A[N x KDIM] @ W[KDIM x 256] ----------------
// One wave computes one 16x16 output tile with V_WMMA_F32_16X16X4_F32.
// 8 waves per block share one N-tile of W staged in LDS (row pad 17 -> no bank conflicts).
// KDIM is compile-time so the main loop is branch-free: 1x global_load_b64 (A) +
// 1x ds_load_2addr_b32 (B) per WMMA.
template <int KDIM>
__global__ void gemm_wmma_kernel(const float* __restrict__ A,
                                 const float* __restrict__ W,
                                 float* __restrict__ out) {
    constexpr int KROUND = (KDIM + 3) & ~3;   // 76 or 256
    constexpr int KMAIN  = KDIM & ~3;         // 72 or 256
    __shared__ float Wt[KROUND * 17];

    const int tid  = threadIdx.x;
    const int lane = tid & 31;
    const int wave = tid >> 5;
    const int col0 = blockIdx.y << 4;
    const int row0 = ((blockIdx.x << 3) + wave) << 4;

    // cooperative stage of W[:, col0:col0+16]; zero-pad rows beyond KDIM
    for (int idx = tid; idx < KROUND * 16; idx += 256) {
        int k = idx >> 4, c = idx & 15;
        Wt[k * 17 + c] = (k < KDIM) ? W[k * CH + col0 + c] : 0.0f;
    }
    __syncthreads();

    const int m  = lane & 15;          // row (A) / col (B) within tile
    const int kh = (lane >> 4) << 1;   // lanes 0-15 -> K+0, lanes 16-31 -> K+2
    const float* arow = A + (size_t)(row0 + m) * KDIM;

    v8f acc = {};
#pragma unroll 4
    for (int k0 = 0; k0 < KMAIN; k0 += 4) {
        const int k = k0 + kh;
        // KDIM even and k even => (row*KDIM + k) even => 8-byte aligned
        v2f a = *(const v2f*)(arow + k);
        v2f b;
        b.x = Wt[k * 17 + m];
        b.y = Wt[(k + 1) * 17 + m];
        acc = __builtin_amdgcn_wmma_f32_16x16x4_f32(
            false, a, false, b, (short)0, acc, false, false);
    }
    if constexpr (KMAIN < KROUND) {
        // tail: clamp addresses, mask values with selects (no EXEC divergence)
        const int k = KMAIN + kh;
        const int i0 = (k     < KDIM) ? k     : (KDIM - 1);
        const int i1 = (k + 1 < KDIM) ? k + 1 : (KDIM - 1);
        float t0 = arow[i0];
        float t1 = arow[i1];
        v2f a, b;
        a.x = (k     < KDIM) ? t0 : 0.0f;
        a.y = (k + 1 < KDIM) ? t1 : 0.0f;
        b.x = Wt[k * 17 + m];              // zero-padded rows
        b.y = Wt[(k + 1) * 17 + m];
        acc = __builtin_amdgcn_wmma_f32_16x16x4_f32(
            false, a, false, b, (short)0, acc, false, false);
    }

    // C/D layout: VGPR r -> M=r (lanes 0-15) / M=8+r (lanes 16-31), N=lane&15
    const int rbase = row0 + ((lane >> 4) << 3);
#pragma unroll
    for (int r = 0; r < 8; ++r) {
        out[(size_t)(rbase + r) * CH + col0 + m] = acc[r];
    }
}

// ---------------- edge aggregation: agg[dst] += hw[src] * dinv[src]*dinv[dst] ----------------
// one wave per edge, 8 channels per lane
__global__ void aggregate_kernel(const int* __restrict__ src, const int* __restrict__ dst,
                                 const float* __restrict__ dinv,
                                 const float* __restrict__ hw,
                                 float* __restrict__ agg, int E) {
    const int lane = threadIdx.x & 31;
    const int e = blockIdx.x * 8 + (threadIdx.x >> 5);
    if (e >= E) return;
    const int s = src[e], d = dst[e];
    const float en = dinv[s] * dinv[d];
    const int base = lane * 8;
    const float4* hp = (const float4*)(hw + (size_t)s * CH + base);
    float4 v0 = hp[0], v1 = hp[1];
    float* ap = agg + (size_t)d * CH + base;
    atomicAdd(ap + 0, v0.x * en);
    atomicAdd(ap + 1, v0.y * en);
    atomicAdd(ap + 2, v0.z * en);
    atomicAdd(ap + 3, v0.w * en);
    atomicAdd(ap + 4, v1.x * en);
    atomicAdd(ap + 5, v1.y * en);
    atomicAdd(ap + 6, v1.z * en);
    atomicAdd(ap + 7, v1.w * en);
}

// ---------------- self-loop + bias + LayerNorm + ReLU + residual (in-place into agg) ----------------
// one wave per node, 8 channels per lane; wave32 shuffle reduction
__global__ void post_kernel(float* __restrict__ agg,           // in: agg, out: h_next
                            const float* __restrict__ hw,
                            const float* __restrict__ hprev,   // residual (nullable)
                            const float* __restrict__ dinv,
                            const float* __restrict__ bias,
                            const float* __restrict__ lg,
                            const float* __restrict__ lb,
                            int N) {
    const int lane = threadIdx.x & 31;
    const int n = blockIdx.x * 8 + (threadIdx.x >> 5);
    if (n >= N) return;
    const float sn = dinv[n] * dinv[n];
    const int base = lane * 8;
    const float4* ap4 = (const float4*)(agg + (size_t)n * CH + base);
    const float4* hp4 = (const float4*)(hw + (size_t)n * CH + base);
    float4 a0 = ap4[0], a1 = ap4[1], h0 = hp4[0], h1 = hp4[1];
    float v[8];
    v[0] = a0.x + h0.x * sn + bias[base + 0];
    v[1] = a0.y + h0.y * sn + bias[base + 1];
    v[2] = a0.z + h0.z * sn + bias[base + 2];
    v[3] = a0.w + h0.w * sn + bias[base + 3];
    v[4] = a1.x + h1.x * sn + bias[base + 4];
    v[5] = a1.y + h1.y * sn + bias[base + 5];
    v[6] = a1.z + h1.z * sn + bias[base + 6];
    v[7] = a1.w + h1.w * sn + bias[base + 7];

    float s = 0.f, s2 = 0.f;
#pragma unroll
    for (int i = 0; i < 8; ++i) { s += v[i]; s2 += v[i] * v[i]; }
#pragma unroll
    for (int off = 16; off > 0; off >>= 1) {
        s  += __shfl_xor(s, off, 32);
        s2 += __shfl_xor(s2, off, 32);
    }
    const float mean = s * (1.0f / CH);
    const float var  = s2 * (1.0f / CH) - mean * mean;
    const float rs   = rsqrtf(var + EPSV);

    float* op = agg + (size_t)n * CH + base;
#pragma unroll
    for (int i = 0; i < 8; ++i) {
        float y = fmaxf((v[i] - mean) * rs * lg[base + i] + lb[base + i], 0.0f);
        if (hprev) y += hprev[(size_t)n * CH + base + i];
        op[i] = y;
    }
}

// ---------------- global mean pool (atomic accumulation) ----------------
__global__ void pool_kernel(const float* __restrict__ h, const int* __restrict__ batch,
                            float* __restrict__ psum, float* __restrict__ pcnt, int N) {
    const int lane = threadIdx.x & 31;
    const int n = blockIdx.x * 8 + (threadIdx.x >> 5);
    if (n >= N) return;
    const int g = batch[n];
    const int base = lane * 8;
    const float4* hp = (const float4*)(h + (size_t)n * CH + base);
    float4 v0 = hp[0], v1 = hp[1];
    float* pp = psum + (size_t)g * CH + base;
    atomicAdd(pp + 0, v0.x);
    atomicAdd(pp + 1, v0.y);
    atomicAdd(pp + 2, v0.z);
    atomicAdd(pp + 3, v0.w);
    atomicAdd(pp + 4, v1.x);
    atomicAdd(pp + 5, v1.y);
    atomicAdd(pp + 6, v1.z);
    atomicAdd(pp + 7, v1.w);
    if (lane == 0) atomicAdd(&pcnt[g], 1.0f);
}

// ---------------- predictor: mean, BN, ReLU, fc1, BN, ReLU, fc2 ----------------
// one block (128 threads) per graph
__global__ void predict_kernel(const float* __restrict__ psum, const float* __restrict__ pcnt,
                               const float* __restrict__ b1g, const float* __restrict__ b1b,
                               const float* __restrict__ b1m, const float* __restrict__ b1v,
                               const float* __restrict__ wf1, const float* __restrict__ bf1,
                               const float* __restrict__ b2g, const float* __restrict__ b2b,
                               const float* __restrict__ b2m, const float* __restrict__ b2v,
                               const float* __restrict__ wf2, const float* __restrict__ bf2,
                               float* __restrict__ out) {
    __shared__ float gv[256];
    __shared__ float red[4];
    const int g = blockIdx.x;
    const int tid = threadIdx.x;
    const float inv_cnt = 1.0f / fmaxf(pcnt[g], 1.0f);
    for (int c = tid; c < 256; c += 128) {
        float v = psum[(size_t)g * CH + c] * inv_cnt;
        v = (v - b1m[c]) * rsqrtf(b1v[c] + EPSV) * b1g[c] + b1b[c];
        gv[c] = fmaxf(v, 0.0f);
    }
    __syncthreads();
    float acc = bf1[tid];
    for (int c = 0; c < 256; ++c) acc += gv[c] * wf1[c * 128 + tid];
    acc = (acc - b2m[tid]) * rsqrtf(b2v[tid] + EPSV) * b2g[tid] + b2b[tid];
    acc = fmaxf(acc, 0.0f);
    float p = acc * wf2[tid];
#pragma unroll
    for (int off = 16; off > 0; off >>= 1) p += __shfl_xor(p, off, 32);
    if ((tid & 31) == 0) red[tid >> 5] = p;
    __syncthreads();
    if (tid == 0) out[g] = red[0] + red[1] + red[2] + red[3] + bf2[0];
}

// ---------------- host side ----------------

extern "C" void kernel_launch(void* const* d_in, const int* in_sizes, int n_in,
                              void* d_out, int out_size, void* d_ws, size_t ws_size,
                              hipStream_t stream) {
    const float* x    = (const float*)d_in[0];
    const int*   ei   = (const int*)d_in[1];
    const int*   bat  = (const int*)d_in[2];
    const float* w0   = (const float*)d_in[3];
    const float* b0   = (const float*)d_in[4];
    const float* wsL  = (const float*)d_in[5];
    const float* bsL  = (const float*)d_in[6];
    const float* lng  = (const float*)d_in[7];
    const float* lnb  = (const float*)d_in[8];
    const float* b1g  = (const float*)d_in[9];
    const float* b1b  = (const float*)d_in[10];
    const float* b1m  = (const float*)d_in[11];
    const float* b1v  = (const float*)d_in[12];
    const float* wf1  = (const float*)d_in[13];
    const float* bf1  = (const float*)d_in[14];
    const float* b2g  = (const float*)d_in[15];
    const float* b2b  = (const float*)d_in[16];
    const float* b2m  = (const float*)d_in[17];
    const float* b2v  = (const float*)d_in[18];
    const float* wf2  = (const float*)d_in[19];
    const float* bf2  = (const float*)d_in[20];
    float* out = (float*)d_out;

    const int* srcI = ei;
    const int* dstI = ei + NE;

    const long long NB = (long long)NND * CH;
    float* bufH = (float*)d_ws;           // h ping
    float* bufG = bufH + NB;              // h pong / agg
    float* hw   = bufG + NB;              // h @ W
    float* dinv = hw + NB;                // degree -> rsqrt
    float* psum = dinv + NND;             // [NG, CH]
    float* pcnt = psum + (long long)NG * CH;

    const dim3 blk256(256);
    const dim3 gemmGrid(NND / 16 / 8, CH / 16);   // (1024, 16)
    const int edgeBlocks = NE / 8;
    const int nodeBlocks = NND / 8;

    // ---- degrees ----
    fill4_kernel<<<(NND / 4 + 255) / 256, blk256, 0, stream>>>((float4*)dinv, 1.0f, NND / 4);
    degree_kernel<<<(NE + 255) / 256, blk256, 0, stream>>>(dstI, dinv, NE);
    rsqrt_kernel<<<(NND + 255) / 256, blk256, 0, stream>>>(dinv, NND);

    // ---- layer 0: x[N,74] -> bufG ----
    gemm_wmma_kernel<74><<<gemmGrid, blk256, 0, stream>>>(x, w0, hw);
    fill4_kernel<<<(unsigned)((NB / 4 + 255) / 256), blk256, 0, stream>>>((float4*)bufG, 0.0f, NB / 4);
    aggregate_kernel<<<edgeBlocks, blk256, 0, stream>>>(srcI, dstI, dinv, hw, bufG, NE);
    post_kernel<<<nodeBlocks, blk256, 0, stream>>>(bufG, hw, nullptr, dinv, b0, lng, lnb, NND);

    // ---- layers 1..3 (ping-pong bufG <-> bufH) ----
    float* hcur = bufG;
    float* hnxt = bufH;
    for (int i = 1; i < 4; ++i) {
        const float* Wi = wsL + (size_t)(i - 1) * CH * CH;
        const float* bi = bsL + (size_t)(i - 1) * CH;
        gemm_wmma_kernel<256><<<gemmGrid, blk256, 0, stream>>>(hcur, Wi, hw);
        fill4_kernel<<<(unsigned)((NB / 4 + 255) / 256), blk256, 0, stream>>>((float4*)hnxt, 0.0f, NB / 4);
        aggregate_kernel<<<edgeBlocks, blk256, 0, stream>>>(srcI, dstI, dinv, hw, hnxt, NE);
        post_kernel<<<nodeBlocks, blk256, 0, stream>>>(hnxt, hw, hcur, dinv, bi,
                                                       lng + (size_t)i * CH, lnb + (size_t)i * CH, NND);
        float* t = hcur; hcur = hnxt; hnxt = t;
    }
    // after 3 swaps, hcur = bufH holds the final node features

    // ---- pooling ----
    fill4_kernel<<<((long long)NG * CH / 4 + 255) / 256, blk256, 0, stream>>>((float4*)psum, 0.0f, (long long)NG * CH / 4);
    fill4_kernel<<<(NG / 4 + 255) / 256, blk256, 0, stream>>>((float4*)pcnt, 0.0f, NG / 4);
    pool_kernel<<<nodeBlocks, blk256, 0, stream>>>(hcur, bat, psum, pcnt, NND);

    // ---- predictor ----
    predict_kernel<<<NG, 128, 0, stream>>>(psum, pcnt, b1g, b1b, b1m, b1v, wf1, bf1,
                                           b2g, b2b, b2m, b2v, wf2, bf2, out);
}